// MALayer_54296976556533
// MI455X (gfx1250) — compile-verified
//
#include <hip/hip_runtime.h>
#include <math.h>

typedef __attribute__((ext_vector_type(2))) float v2f;
typedef __attribute__((ext_vector_type(8))) float v8f;

#define IN_DIM 64

// ---------------------------------------------------------------------------
// Zero the output accumulator (z is accumulated with atomics each launch).
// ---------------------------------------------------------------------------
__global__ __launch_bounds__(256) void zero_kernel(float* __restrict__ p, int n) {
  int i = blockIdx.x * blockDim.x + threadIdx.x;
  if (i < n) p[i] = 0.0f;
}

// ---------------------------------------------------------------------------
// Gate scores via V_WMMA_F32_16X16X4_F32 (full-precision f32 WMMA).
// One wave computes s_dst and s_src for a 16-node tile:
//   A (16x4)  = tile of h      (lane M = lane&15; lanes 0-15 carry K=k,k+1,
//                               lanes 16-31 carry K=k+2,k+3 per ISA layout)
//   B (4x16)  = column N=0 : w_dst (gate_w[0..63])
//               column N=1 : w_src (gate_w[64..127]), other columns 0
//   C/D 16x16 = column 0 -> s_dst, column 1 -> s_src
// 16 chained WMMAs accumulate over K=64. B-masking is branch-free so EXEC
// is all-1s at every WMMA (hardware requirement).
// D layout: lane L holds column N=L&15; VGPR r holds row M = r + 8*(L>>4).
// ---------------------------------------------------------------------------
__global__ __launch_bounds__(256) void gate_scores_wmma(
    const float* __restrict__ h, const float* __restrict__ gate_w,
    float* __restrict__ sdst, float* __restrict__ ssrc, int n_nodes) {
  const int lane  = threadIdx.x & 31;
  const int tile  = blockIdx.x * (blockDim.x >> 5) + (threadIdx.x >> 5);
  const int base  = tile * 16;
  if (base >= n_nodes) return;              // whole-wave uniform exit only

  const int M     = lane & 15;
  const int khalf = lane >> 4;              // 0: lanes 0-15, 1: lanes 16-31

  int node = base + M;
  if (node >= n_nodes) node = n_nodes - 1;  // clamp: keep EXEC all-1s
  const float* __restrict__ hrow = h + (size_t)node * IN_DIM;

  const float  bmask = (M < 2) ? 1.0f : 0.0f;           // zero cols N>=2
  const float* __restrict__ wcol = gate_w + ((M == 1) ? IN_DIM : 0);

  v8f c = {};
#pragma unroll
  for (int k0 = 0; k0 < IN_DIM; k0 += 4) {
    const int kk = k0 + 2 * khalf;
    v2f a, b;
    a.x = hrow[kk];
    a.y = hrow[kk + 1];
    b.x = wcol[kk] * bmask;
    b.y = wcol[kk + 1] * bmask;
    // 8 args: (neg_a, A, neg_b, B, c_mod, C, reuse_a, reuse_b)
    c = __builtin_amdgcn_wmma_f32_16x16x4_f32(false, a, false, b, (short)0, c,
                                              false, false);
  }

  // Column N=0 lives in lanes 0/16 (s_dst), N=1 in lanes 1/17 (s_src).
  if (M < 2) {
    float* __restrict__ outp = (M == 0) ? sdst : ssrc;
    const int row0 = base + 8 * khalf;
#pragma unroll
    for (int r = 0; r < 8; ++r) {
      const int row = row0 + r;
      if (row < n_nodes) outp[row] = c[r];
    }
  }
}

// ---------------------------------------------------------------------------
// Edge scatter: one wave per edge; lane handles features 2*lane, 2*lane+1.
// h and z (12.8 MB each) are L2-resident on MI455X (192 MB L2), so the
// gather + hardware global_atomic_add_f32 RMW both stay on-chip.
// ---------------------------------------------------------------------------
__global__ __launch_bounds__(256) void edge_scatter(
    const float* __restrict__ h, const float* __restrict__ dvec,
    const int* __restrict__ src, const int* __restrict__ dst,
    const float* __restrict__ sdst, const float* __restrict__ ssrc,
    const float* __restrict__ gate_b, float* __restrict__ z, int n_edges) {
  const int edge = (blockIdx.x * blockDim.x + threadIdx.x) >> 5;
  const int lane = threadIdx.x & 31;
  if (edge >= n_edges) return;

  const int s = src[edge];
  const int t = dst[edge];
  const float g    = tanhf(sdst[t] + ssrc[s] + gate_b[0]);
  const float coef = g * dvec[t] * dvec[s];

  const float2 hv =
      *reinterpret_cast<const float2*>(h + (size_t)s * IN_DIM + 2 * lane);
  float* zp = z + (size_t)t * IN_DIM + 2 * lane;
  unsafeAtomicAdd(zp,     hv.x * coef);   // -> global_atomic_add_f32
  unsafeAtomicAdd(zp + 1, hv.y * coef);
}

// ---------------------------------------------------------------------------
extern "C" void kernel_launch(void* const* d_in, const int* in_sizes, int n_in,
                              void* d_out, int out_size, void* d_ws,
                              size_t ws_size, hipStream_t stream) {
  const float* h      = (const float*)d_in[0];
  const float* dvec   = (const float*)d_in[1];
  const int*   src    = (const int*)d_in[2];
  const int*   dst    = (const int*)d_in[3];
  const float* gate_w = (const float*)d_in[4];
  const float* gate_b = (const float*)d_in[5];
  float* z = (float*)d_out;

  const int n_nodes = in_sizes[1];   // d has one entry per node
  const int n_edges = in_sizes[2];   // src has one entry per edge

  float* sdst = (float*)d_ws;        // n_nodes floats
  float* ssrc = sdst + n_nodes;      // n_nodes floats  (total 400 KB @ 50 K)

  // 1) zero output accumulator
  {
    const int n = n_nodes * IN_DIM;
    zero_kernel<<<(n + 255) / 256, 256, 0, stream>>>(z, n);
  }

  // 2) gate scores via f32 WMMA: 8 waves/block, one 16-node tile per wave
  {
    const int tiles  = (n_nodes + 15) / 16;
    const int blocks = (tiles + 7) / 8;
    gate_scores_wmma<<<blocks, 256, 0, stream>>>(h, gate_w, sdst, ssrc,
                                                 n_nodes);
  }

  // 3) edge scatter with hardware f32 atomics (one wave per edge)
  {
    const int waves_per_block = 8;
    const int blocks = (n_edges + waves_per_block - 1) / waves_per_block;
    edge_scatter<<<blocks, waves_per_block * 32, 0, stream>>>(
        h, dvec, src, dst, sdst, ssrc, gate_b, z, n_edges);
  }
}